// Projection_19628000543291
// MI455X (gfx1250) — compile-verified
//
#include <hip/hip_runtime.h>
#include <hip/hip_bf16.h>
#include <math.h>

// ---------------------------------------------------------------------------
// Sizes from the reference
// ---------------------------------------------------------------------------
#define BATCH       256
#define HWPX        (56 * 56)       // 3136 elements per CAM plane
#define NCH         31              // 6 + 10 + 15 channels pooled
#define OD          128
#define NH          8
#define HD          16
#define NTRIP       100
#define LSEQ        200
#define D_STATE     16
#define D_INNER     2
#define EPSBN       1e-5f

// Workspace layout (float offsets); total = 307200 floats = 1.2 MB
#define OFF_G       0               // (256,31) pooled means
#define OFF_q       8192            // (256,128) pre-projection q
#define OFF_k       40960
#define OFF_v       73728
#define OFF_Q       106496          // (256,128) post q_w
#define OFF_K       139264
#define OFF_V       172032
#define OFF_FEAT    204800          // (256,200) concat(cam_X, Xn)
#define OFF_MFEAT   256000          // (256,200) mamba output

typedef __attribute__((ext_vector_type(2))) float v2f;
typedef __attribute__((ext_vector_type(8))) float v8f;

__device__ __forceinline__ float bn_f(float x, float m, float v, float g, float b) {
    return (x - m) * rsqrtf(v + EPSBN) * g + b;
}
__device__ __forceinline__ float elu_f(float x)  { return x > 0.f ? x : expf(x) - 1.f; }
__device__ __forceinline__ float silu_f(float x) { return x / (1.f + expf(-x)); }
__device__ __forceinline__ float softplus_f(float x) { return x > 20.f ? x : log1pf(expf(x)); }

// ---------------------------------------------------------------------------
// Kernel 1: fused global average pooling of cam_i/cam_v/cam_t.
// One block per (b, merged-channel) plane; 99.6 MB streamed at HBM rate.
// g[b*31 + c]: c in [0,6) = gi, [6,16) = gv, [16,31) = gt.
// ---------------------------------------------------------------------------
__global__ void __launch_bounds__(128)
pool_kernel(const float* __restrict__ ci, const float* __restrict__ cv,
            const float* __restrict__ ct, float* __restrict__ g) {
    int p = blockIdx.x;                 // 0 .. 256*31-1
    int b = p / NCH;
    int c = p - b * NCH;
    const float* src;
    if (c < 6)        src = ci + (size_t)(b * 6  +  c)       * HWPX;
    else if (c < 16)  src = cv + (size_t)(b * 10 + (c - 6))  * HWPX;
    else              src = ct + (size_t)(b * 15 + (c - 16)) * HWPX;

    const float4* s4 = (const float4*)src;   // 3136 = 784 * 4, 16B aligned
    float acc = 0.f;
    for (int i = threadIdx.x; i < 784; i += 128) {
        __builtin_prefetch(s4 + i + 128, 0, 0);   // -> global_prefetch_b8 (speculative)
        float4 x = s4[i];
        acc += x.x + x.y + x.z + x.w;
    }
    __shared__ float red[128];
    red[threadIdx.x] = acc;
    __syncthreads();
    for (int off = 64; off > 0; off >>= 1) {
        if (threadIdx.x < off) red[threadIdx.x] += red[threadIdx.x + off];
        __syncthreads();
    }
    if (threadIdx.x == 0) g[p] = red[0] * (1.f / (float)HWPX);
}

// ---------------------------------------------------------------------------
// Kernel 2: q/k/v = elu(bn(g @ key_w.T + key_b)).  Block = one batch row.
// ---------------------------------------------------------------------------
__global__ void __launch_bounds__(128)
prekey_kernel(const float* __restrict__ g,
              const float* __restrict__ ikw, const float* __restrict__ ikb,
              const float* __restrict__ vkw, const float* __restrict__ vkb,
              const float* __restrict__ tkw, const float* __restrict__ tkb,
              const float* __restrict__ g1, const float* __restrict__ b1,
              const float* __restrict__ m1, const float* __restrict__ v1,
              const float* __restrict__ g2, const float* __restrict__ b2,
              const float* __restrict__ m2, const float* __restrict__ v2,
              const float* __restrict__ g3, const float* __restrict__ b3,
              const float* __restrict__ m3, const float* __restrict__ v3,
              float* __restrict__ qo, float* __restrict__ ko, float* __restrict__ vo) {
    __shared__ float gs[NCH];
    int b = blockIdx.x, o = threadIdx.x;
    if (o < NCH) gs[o] = g[b * NCH + o];
    __syncthreads();
    float aq = ikb[o];
    for (int c = 0; c < 6;  ++c) aq += gs[c]      * ikw[o * 6  + c];
    float ak = vkb[o];
    for (int c = 0; c < 10; ++c) ak += gs[6 + c]  * vkw[o * 10 + c];
    float av = tkb[o];
    for (int c = 0; c < 15; ++c) av += gs[16 + c] * tkw[o * 15 + c];
    qo[b * OD + o] = elu_f(bn_f(aq, m1[o], v1[o], g1[o], b1[o]));
    ko[b * OD + o] = elu_f(bn_f(ak, m2[o], v2[o], g2[o], b2[o]));
    vo[b * OD + o] = elu_f(bn_f(av, m3[o], v3[o], g3[o], b3[o]));
}

// ---------------------------------------------------------------------------
// Kernel 3: Q/K/V = x @ w.T + bias via V_WMMA_F32_16X16X4_F32 (full f32).
// One wave per 16x16 output tile, K=128 -> 32 WMMA steps.
// A layout (16x4 f32): lanes 0-15 K={0,1}, lanes 16-31 K={2,3}; row M = lane&15.
// B layout (4x16 f32): VGPR0 K={0|2}, VGPR1 K={1|3}; col N = lane&15.
// D layout: VGPR i -> M = i + 8*(lane>=16), N = lane&15.
// EXEC all-ones: 32-thread blocks, no divergence before WMMA.
// ---------------------------------------------------------------------------
__global__ void __launch_bounds__(32)
wmma_proj_kernel(const float* __restrict__ q,  const float* __restrict__ k,
                 const float* __restrict__ v,
                 const float* __restrict__ qw, const float* __restrict__ kw,
                 const float* __restrict__ vw,
                 const float* __restrict__ qb, const float* __restrict__ kb,
                 const float* __restrict__ vb,
                 float* __restrict__ Q, float* __restrict__ K, float* __restrict__ V) {
    const float *src, *w, *bias;
    float* dst;
    if (blockIdx.z == 0)      { src = q; w = qw; bias = qb; dst = Q; }
    else if (blockIdx.z == 1) { src = k; w = kw; bias = kb; dst = K; }
    else                      { src = v; w = vw; bias = vb; dst = V; }

    const int n0   = blockIdx.x * 16;
    const int m0   = blockIdx.y * 16;
    const int lane = threadIdx.x;
    const int half = lane >> 4;      // 0: lanes 0-15, 1: lanes 16-31
    const int l15  = lane & 15;

    v8f c = {};
    for (int k0 = 0; k0 < OD; k0 += 4) {
        // A[m][kk] = src[m*OD + kk]
        const float* ap = src + (size_t)(m0 + l15) * OD + k0 + 2 * half;
        v2f a; a.x = ap[0]; a.y = ap[1];
        // B[kk][n] = w[n*OD + kk]  (D = src @ w.T)
        const float* bp = w + (size_t)(n0 + l15) * OD + k0 + 2 * half;
        v2f bb; bb.x = bp[0]; bb.y = bp[1];
        c = __builtin_amdgcn_wmma_f32_16x16x4_f32(
                /*neg_a=*/false, a, /*neg_b=*/false, bb,
                /*c_mod=*/(short)0, c, /*reuse_a=*/false, /*reuse_b=*/false);
    }
    const float bn = bias[n0 + l15];
    #pragma unroll
    for (int i = 0; i < 8; ++i) {
        int row = m0 + i + half * 8;
        dst[(size_t)row * OD + n0 + l15] = c[i] + bn;
    }
}

// ---------------------------------------------------------------------------
// Kernel 4: per-row cross attention + o_w projection + Xn, writes feat(256,200)
// ---------------------------------------------------------------------------
__global__ void __launch_bounds__(128)
attn_kernel(const float* __restrict__ Q, const float* __restrict__ K,
            const float* __restrict__ V,
            const float* __restrict__ ow, const float* __restrict__ ob,
            const float* __restrict__ X,
            const float* __restrict__ g4, const float* __restrict__ b4,
            const float* __restrict__ m4, const float* __restrict__ v4,
            float* __restrict__ feat) {
    int b = blockIdx.x, t = threadIdx.x;
    __shared__ float sQ[OD], sK[OD], sV[OD], sS[NH * NH], sAtt[OD];
    sQ[t] = Q[b * OD + t];
    sK[t] = K[b * OD + t];
    sV[t] = V[b * OD + t];
    __syncthreads();

    if (t < NH * NH) {                       // scores[h][g]
        int h = t >> 3, gg = t & 7;
        float acc = 0.f;
        for (int d = 0; d < HD; ++d) acc += sQ[h * HD + d] * sK[gg * HD + d];
        sS[t] = acc * 0.25f;                 // 1/sqrt(16)
    }
    __syncthreads();
    if (t < NH) {                            // softmax over g (rows of 8)
        float mx = -1e30f;
        for (int gg = 0; gg < NH; ++gg) mx = fmaxf(mx, sS[t * NH + gg]);
        float sum = 0.f;
        for (int gg = 0; gg < NH; ++gg) { float e = expf(sS[t * NH + gg] - mx); sS[t * NH + gg] = e; sum += e; }
        float inv = 1.f / sum;
        for (int gg = 0; gg < NH; ++gg) sS[t * NH + gg] *= inv;
    }
    __syncthreads();
    {                                        // att = probs @ V
        int h = t >> 4, d = t & 15;
        float acc = 0.f;
        for (int gg = 0; gg < NH; ++gg) acc += sS[h * NH + gg] * sV[gg * HD + d];
        sAtt[t] = acc;
    }
    __syncthreads();
    if (t < NTRIP) {                         // cam_X = att @ o_w.T + o_b ; Xn
        float acc = ob[t];
        for (int j = 0; j < OD; ++j) acc += sAtt[j] * ow[t * OD + j];
        feat[b * LSEQ + t] = acc;
        float xn = elu_f(bn_f(X[b * NTRIP + t], m4[t], v4[t], g4[t], b4[t]));
        feat[b * LSEQ + NTRIP + t] = xn;
    }
}

// ---------------------------------------------------------------------------
// Kernel 5: selective-scan mamba. One wave32 per batch row;
// lane = d*16 + s over the (D_INNER=2) x (D_STATE=16) state — exact fit.
// Sequential over L=200; per-step reductions via __shfl_xor butterflies.
// ---------------------------------------------------------------------------
__global__ void __launch_bounds__(32)
mamba_kernel(const float* __restrict__ feat,
             const float* __restrict__ ipw,   // in_proj_w (4,1)
             const float* __restrict__ cw,    // conv_w (2,1,4)
             const float* __restrict__ cb,    // conv_b (2)
             const float* __restrict__ xpw,   // x_proj_w (33,2)
             const float* __restrict__ dtw,   // dt_proj_w (2,1)
             const float* __restrict__ dtb,   // dt_proj_b (2)
             const float* __restrict__ alog,  // A_log (2,16)
             const float* __restrict__ Dpv,   // Dp (2)
             const float* __restrict__ opw,   // out_proj_w (1,2)
             float* __restrict__ mfeat) {
    const int b    = blockIdx.x;
    const int lane = threadIdx.x;
    const int d    = lane >> 4;
    const int s    = lane & 15;

    const float xp00 = xpw[0],                 xp01 = xpw[1];
    const float xpB0 = xpw[(1 + s) * 2 + 0],   xpB1 = xpw[(1 + s) * 2 + 1];
    const float xpC0 = xpw[(17 + s) * 2 + 0],  xpC1 = xpw[(17 + s) * 2 + 1];
    const float A_ds = -expf(alog[d * D_STATE + s]);
    const float dtw_d = dtw[d], dtb_d = dtb[d];
    const float ipw0 = ipw[0], ipw1 = ipw[1];
    const float ipwz_d = ipw[2 + d];
    float cw0[4], cw1[4];
    #pragma unroll
    for (int j = 0; j < 4; ++j) { cw0[j] = cw[j]; cw1[j] = cw[4 + j]; }
    const float cb0 = cb[0], cb1 = cb[1];
    const float Dp_d = Dpv[d], ow_d = opw[d];

    const float* frow = feat + (size_t)b * LSEQ;
    float u0 = 0.f, u1 = 0.f, u2 = 0.f;   // causal conv window (pad 3)
    float h = 0.f;                        // lane-private SSM state h[d][s]

    for (int t = 0; t < LSEQ; ++t) {
        const float u3 = frow[t];
        // depthwise causal conv (fold in_proj: x_pre[.,d] = u * ipw[d])
        const float c0 = u0 * cw0[0] + u1 * cw0[1] + u2 * cw0[2] + u3 * cw0[3];
        const float c1 = u0 * cw1[0] + u1 * cw1[1] + u2 * cw1[2] + u3 * cw1[3];
        const float x0 = silu_f(cb0 + ipw0 * c0);
        const float x1 = silu_f(cb1 + ipw1 * c1);
        const float xd = d ? x1 : x0;

        const float dtr  = x0 * xp00 + x1 * xp01;
        const float dt_d = softplus_f(dtr * dtw_d + dtb_d);
        const float Bm   = x0 * xpB0 + x1 * xpB1;
        const float Cm   = x0 * xpC0 + x1 * xpC1;

        h = expf(dt_d * A_ds) * h + dt_d * Bm * xd;

        float contrib = h * Cm;                 // y_d = sum_s h*C, groups of 16
        contrib += __shfl_xor(contrib, 1, 16);
        contrib += __shfl_xor(contrib, 2, 16);
        contrib += __shfl_xor(contrib, 4, 16);
        contrib += __shfl_xor(contrib, 8, 16);

        float y = contrib + xd * Dp_d;
        y *= silu_f(u3 * ipwz_d);               // gate by silu(z)

        float tot = y * ow_d;                   // out_proj: y0*w0 + y1*w1
        tot += __shfl_xor(tot, 16, 32);
        if (lane == 0) mfeat[(size_t)b * LSEQ + t] = tot;

        u0 = u1; u1 = u2; u2 = u3;
    }
}

// ---------------------------------------------------------------------------
// Kernel 6: out = mfeat @ shape_w.T + shape_b   (256,200)x(200,100)
// ---------------------------------------------------------------------------
__global__ void __launch_bounds__(128)
final_kernel(const float* __restrict__ mf, const float* __restrict__ sw,
             const float* __restrict__ sb, float* __restrict__ out) {
    int b = blockIdx.x, t = threadIdx.x;
    __shared__ float row[LSEQ];
    row[t] = mf[(size_t)b * LSEQ + t];
    if (t < LSEQ - 128) row[128 + t] = mf[(size_t)b * LSEQ + 128 + t];
    __syncthreads();
    if (t < NTRIP) {
        float acc = sb[t];
        for (int j = 0; j < LSEQ; ++j) acc += row[j] * sw[t * LSEQ + j];
        out[(size_t)b * NTRIP + t] = acc;
    }
}

// ---------------------------------------------------------------------------
// Launch
// ---------------------------------------------------------------------------
extern "C" void kernel_launch(void* const* d_in, const int* in_sizes, int n_in,
                              void* d_out, int out_size, void* d_ws, size_t ws_size,
                              hipStream_t stream) {
    // d_in order = setup_inputs() dict order, params flattened in insertion order
    const float* cam_i = (const float*)d_in[0];
    const float* cam_v = (const float*)d_in[1];
    const float* cam_t = (const float*)d_in[2];
    const float* X     = (const float*)d_in[3];
    const float* ikw = (const float*)d_in[4];   const float* ikb = (const float*)d_in[5];
    const float* vkw = (const float*)d_in[6];   const float* vkb = (const float*)d_in[7];
    const float* tkw = (const float*)d_in[8];   const float* tkb = (const float*)d_in[9];
    const float* bn1g = (const float*)d_in[10]; const float* bn1b = (const float*)d_in[11];
    const float* bn1m = (const float*)d_in[12]; const float* bn1v = (const float*)d_in[13];
    const float* bn2g = (const float*)d_in[14]; const float* bn2b = (const float*)d_in[15];
    const float* bn2m = (const float*)d_in[16]; const float* bn2v = (const float*)d_in[17];
    const float* bn3g = (const float*)d_in[18]; const float* bn3b = (const float*)d_in[19];
    const float* bn3m = (const float*)d_in[20]; const float* bn3v = (const float*)d_in[21];
    const float* bn4g = (const float*)d_in[22]; const float* bn4b = (const float*)d_in[23];
    const float* bn4m = (const float*)d_in[24]; const float* bn4v = (const float*)d_in[25];
    const float* qw = (const float*)d_in[26];   const float* qb = (const float*)d_in[27];
    const float* kw = (const float*)d_in[28];   const float* kb = (const float*)d_in[29];
    const float* vw = (const float*)d_in[30];   const float* vb = (const float*)d_in[31];
    const float* ow = (const float*)d_in[32];   const float* ob = (const float*)d_in[33];
    const float* ipw = (const float*)d_in[34];
    const float* cw  = (const float*)d_in[35];  const float* cb  = (const float*)d_in[36];
    const float* xpw = (const float*)d_in[37];
    const float* dtw = (const float*)d_in[38];  const float* dtb = (const float*)d_in[39];
    const float* alog = (const float*)d_in[40]; const float* Dpv = (const float*)d_in[41];
    const float* opw = (const float*)d_in[42];
    const float* sw  = (const float*)d_in[43];  const float* sb  = (const float*)d_in[44];

    float* ws = (float*)d_ws;
    float* g   = ws + OFF_G;
    float* q   = ws + OFF_q;  float* k  = ws + OFF_k;  float* v  = ws + OFF_v;
    float* Q   = ws + OFF_Q;  float* K  = ws + OFF_K;  float* V  = ws + OFF_V;
    float* feat  = ws + OFF_FEAT;
    float* mfeat = ws + OFF_MFEAT;

    // 1. bandwidth-dominant pooling (99.6 MB streamed)
    pool_kernel<<<BATCH * NCH, 128, 0, stream>>>(cam_i, cam_v, cam_t, g);
    // 2. key projections + bn + elu
    prekey_kernel<<<BATCH, 128, 0, stream>>>(g, ikw, ikb, vkw, vkb, tkw, tkb,
                                             bn1g, bn1b, bn1m, bn1v,
                                             bn2g, bn2b, bn2m, bn2v,
                                             bn3g, bn3b, bn3m, bn3v, q, k, v);
    // 3. f32 WMMA GEMMs for Q/K/V
    wmma_proj_kernel<<<dim3(OD / 16, BATCH / 16, 3), 32, 0, stream>>>(
        q, k, v, qw, kw, vw, qb, kb, vb, Q, K, V);
    // 4. attention + concat -> feat
    attn_kernel<<<BATCH, 128, 0, stream>>>(Q, K, V, ow, ob, X,
                                           bn4g, bn4b, bn4m, bn4v, feat);
    // 5. wave32-per-row selective scan
    mamba_kernel<<<BATCH, 32, 0, stream>>>(feat, ipw, cw, cb, xpw, dtw, dtb,
                                           alog, Dpv, opw, mfeat);
    // 6. final projection -> d_out (256,100) f32
    final_kernel<<<BATCH, 128, 0, stream>>>(mfeat, sw, sb, (float*)d_out);
}